// UFPNet_39049842655642
// MI455X (gfx1250) — compile-verified
//
#include <hip/hip_runtime.h>
#include <math.h>

typedef float v2f __attribute__((ext_vector_type(2)));
typedef float v8f __attribute__((ext_vector_type(8)));

#define BATCH  16
#define LMASK  25
#define NFULL  256
#define NLR    50
#define NPAD   64
#define TSWEEP 3
#define SITER  8
#define PLF    (NFULL*NFULL)   /* 65536 plane elems, full-res */
#define PLP    (NPAD*NPAD)     /* 4096 plane elems, padded low-res */

// ---------------------------------------------------------------------------
// WMMA f32 16x16x4 wrapper:  D = A(16x4) * B(4x16) + C(16x16)
// ---------------------------------------------------------------------------
__device__ __forceinline__ v8f wmma4(v2f a, v2f b, v8f c) {
  return __builtin_amdgcn_wmma_f32_16x16x4_f32(
      /*neg_a=*/false, a, /*neg_b=*/false, b,
      /*c_mod=*/(short)0, c, /*reuse_a=*/false, /*reuse_b=*/false);
}

// ---------------------------------------------------------------------------
// Register-blocked, LDS-staged complex GEMM (used by the AFFP batched FFTs).
//   TRANSB == 0 :  C[b] = A * B[b]        (B slab read row-major:   B[k][c])
//   TRANSB == 1 :  C[b] = A[b] * B^T      (B slab read transposed:  B[c][k])
// Complex fields are planar ([real plane][imag plane]); batched operands use
// stride 2*Np*Np, shared DFT matrices stride 0.
// Block = 128 threads (4 waves), 64x64 output region; each wave computes a
// 32x32 sub-block as 2x2 WMMA tiles from K=16 LDS slabs (stride-17 pad =>
// conflict-free ds reads for the wave32 A(16x4)/B(4x16) fragment layouts).
// ---------------------------------------------------------------------------
template <int TRANSB>
__global__ __launch_bounds__(128)
void k_cgemm(const float* __restrict__ A, const float* __restrict__ B,
             float* __restrict__ C, int Np,
             unsigned long long sA, unsigned long long sB) {
  __shared__ float As[2][64][17];
  __shared__ float Bs[2][64][17];

  const int t = threadIdx.x;
  const int lane = t & 31, w = t >> 5;
  const int half = lane >> 4, lr = lane & 15;
  const int tn0 = blockIdx.x * 64, tm0 = blockIdx.y * 64;
  const int b = blockIdx.z;
  const int plane = Np * Np;

  const float* Ar = A + (size_t)b * sA;
  const float* Ai = Ar + plane;
  const float* Br = B + (size_t)b * sB;
  const float* Bi = Br + plane;
  float*       Cr = C + (size_t)b * 2 * plane;
  float*       Ci = Cr + plane;

  const int wm = (w >> 1) * 32, wn = (w & 1) * 32;

  v8f acr00 = {}, acr01 = {}, acr10 = {}, acr11 = {};
  v8f aci00 = {}, aci01 = {}, aci10 = {}, aci11 = {};

  for (int ks = 0; ks < Np; ks += 16) {
    for (int idx = t; idx < 64 * 16; idx += 128) {
      const int r = idx >> 4, k = idx & 15;
      As[0][r][k] = Ar[(tm0 + r) * Np + ks + k];
      As[1][r][k] = Ai[(tm0 + r) * Np + ks + k];
    }
    if (TRANSB) {
      for (int idx = t; idx < 64 * 16; idx += 128) {
        const int c = idx >> 4, k = idx & 15;
        Bs[0][c][k] = Br[(tn0 + c) * Np + ks + k];
        Bs[1][c][k] = Bi[(tn0 + c) * Np + ks + k];
      }
    } else {
      for (int idx = t; idx < 64 * 16; idx += 128) {
        const int k = idx >> 6, c = idx & 63;
        Bs[0][c][k] = Br[(ks + k) * Np + tn0 + c];
        Bs[1][c][k] = Bi[(ks + k) * Np + tn0 + c];
      }
    }
    if (ks + 16 < Np) {
      __builtin_prefetch(&Ar[(tm0 + (t >> 1)) * Np + ks + 16], 0, 1);
    }
    __syncthreads();

#pragma unroll
    for (int kk = 0; kk < 16; kk += 4) {
      const int kc = kk + 2 * half;
      v2f a0r, a0i, a1r, a1i, b0r, b0i, b1r, b1i, n0i, n1i;
      a0r.x = As[0][wm + lr][kc];       a0r.y = As[0][wm + lr][kc + 1];
      a0i.x = As[1][wm + lr][kc];       a0i.y = As[1][wm + lr][kc + 1];
      a1r.x = As[0][wm + 16 + lr][kc];  a1r.y = As[0][wm + 16 + lr][kc + 1];
      a1i.x = As[1][wm + 16 + lr][kc];  a1i.y = As[1][wm + 16 + lr][kc + 1];
      b0r.x = Bs[0][wn + lr][kc];       b0r.y = Bs[0][wn + lr][kc + 1];
      b0i.x = Bs[1][wn + lr][kc];       b0i.y = Bs[1][wn + lr][kc + 1];
      b1r.x = Bs[0][wn + 16 + lr][kc];  b1r.y = Bs[0][wn + 16 + lr][kc + 1];
      b1i.x = Bs[1][wn + 16 + lr][kc];  b1i.y = Bs[1][wn + 16 + lr][kc + 1];
      n0i.x = -a0i.x; n0i.y = -a0i.y;
      n1i.x = -a1i.x; n1i.y = -a1i.y;

      acr00 = wmma4(a0r, b0r, acr00);  acr00 = wmma4(n0i, b0i, acr00);
      aci00 = wmma4(a0r, b0i, aci00);  aci00 = wmma4(a0i, b0r, aci00);
      acr01 = wmma4(a0r, b1r, acr01);  acr01 = wmma4(n0i, b1i, acr01);
      aci01 = wmma4(a0r, b1i, aci01);  aci01 = wmma4(a0i, b1r, aci01);
      acr10 = wmma4(a1r, b0r, acr10);  acr10 = wmma4(n1i, b0i, acr10);
      aci10 = wmma4(a1r, b0i, aci10);  aci10 = wmma4(a1i, b0r, aci10);
      acr11 = wmma4(a1r, b1r, acr11);  acr11 = wmma4(n1i, b1i, acr11);
      aci11 = wmma4(a1r, b1i, aci11);  aci11 = wmma4(a1i, b1r, aci11);
    }
    __syncthreads();
  }

  const int rm0 = tm0 + wm, cn0 = tn0 + wn;
#pragma unroll
  for (int r = 0; r < 8; ++r) {
    const int row0 = rm0 + r + 8 * half;
    const int row1 = row0 + 16;
    Cr[row0 * Np + cn0 + lr]      = acr00[r];
    Cr[row0 * Np + cn0 + 16 + lr] = acr01[r];
    Cr[row1 * Np + cn0 + lr]      = acr10[r];
    Cr[row1 * Np + cn0 + 16 + lr] = acr11[r];
    Ci[row0 * Np + cn0 + lr]      = aci00[r];
    Ci[row0 * Np + cn0 + 16 + lr] = aci01[r];
    Ci[row1 * Np + cn0 + lr]      = aci10[r];
    Ci[row1 * Np + cn0 + 16 + lr] = aci11[r];
  }
}

// ---------------------------------------------------------------------------
// Fused fp_basic step (one launch per (t,l) sweep step).
// One block per batch image, 256 threads = 8 waves; each wave owns a 32x16
// output strip (2x1 WMMA tiles).  The whole step runs in LDS:
//   phase0: WA = (crop(OFT)*CTF)^T            [n][k] = B-fragment layout
//   phase1: WB = GS * P                        (A: global GS, B: WA)
//   phase2: WA = WB * GS^T        -> low       (A: WB row-major, B: global GS)
//   phase3: WB = (sqrtY * low/|low|)^T         transpose fused into pointwise
//   phase4: WA = FS * low_new                  (A: global FS, B: WB)
//   phase5: part = WA * FS^T; OFT[crop] = (1-ctf)*patch + ctf*part  (fused)
// LDS rows padded to 68 floats: bank = (4*idx + k) mod 64, conflict-free for
// both halves of the wave32 fragment reads.
// ---------------------------------------------------------------------------
__global__ __launch_bounds__(256)
void k_fp_step(float* __restrict__ OFT, const float* __restrict__ GS,
               const float* __restrict__ FS, const float* __restrict__ CTF,
               const float* __restrict__ Y, const int* __restrict__ Masks,
               int l) {
  __shared__ float WA[2][64][68];
  __shared__ float WB[2][64][68];

  const int t = threadIdx.x;
  const int lane = t & 31, w = t >> 5;
  const int half = lane >> 4, lr = lane & 15;
  const int b = blockIdx.x;
  const int r0 = Masks[2 * l] - 1, c0 = Masks[2 * l + 1] - 1;

  float* Or = OFT + (size_t)b * 2 * PLF;
  float* Oi = Or + PLF;
  const float* GSr = GS;  const float* GSi = GS + PLP;
  const float* FSr = FS;  const float* FSi = FS + PLP;

  const int wm = (w >> 2) * 32;   // 0 or 32
  const int wn = (w & 3) * 16;    // 0,16,32,48

  // ---- phase 0: WA[c][m] = patch[m][c] * CTF ----
  for (int idx = t; idx < PLP; idx += 256) {
    const int m = idx >> 6, c = idx & 63;
    float vr = 0.f, vi = 0.f;
    if (m < NLR && c < NLR) {
      const float ct = CTF[m * NLR + c];
      vr = Or[(r0 + m) * NFULL + c0 + c] * ct;
      vi = Oi[(r0 + m) * NFULL + c0 + c] * ct;
    }
    WA[0][c][m] = vr;
    WA[1][c][m] = vi;
  }
  __syncthreads();

  // ---- phase 1: WB = GS * P ----
  {
    v8f cr0 = {}, cr1 = {}, ci0 = {}, ci1 = {};
#pragma unroll 4
    for (int kk = 0; kk < NPAD; kk += 4) {
      const int kc = kk + 2 * half;
      v2f a0r, a0i, a1r, a1i, br, bi, n0, n1;
      a0r.x = GSr[(wm + lr) * NPAD + kc];       a0r.y = GSr[(wm + lr) * NPAD + kc + 1];
      a0i.x = GSi[(wm + lr) * NPAD + kc];       a0i.y = GSi[(wm + lr) * NPAD + kc + 1];
      a1r.x = GSr[(wm + 16 + lr) * NPAD + kc];  a1r.y = GSr[(wm + 16 + lr) * NPAD + kc + 1];
      a1i.x = GSi[(wm + 16 + lr) * NPAD + kc];  a1i.y = GSi[(wm + 16 + lr) * NPAD + kc + 1];
      br.x = WA[0][wn + lr][kc];  br.y = WA[0][wn + lr][kc + 1];
      bi.x = WA[1][wn + lr][kc];  bi.y = WA[1][wn + lr][kc + 1];
      n0.x = -a0i.x; n0.y = -a0i.y; n1.x = -a1i.x; n1.y = -a1i.y;
      cr0 = wmma4(a0r, br, cr0);  cr0 = wmma4(n0, bi, cr0);
      ci0 = wmma4(a0r, bi, ci0);  ci0 = wmma4(a0i, br, ci0);
      cr1 = wmma4(a1r, br, cr1);  cr1 = wmma4(n1, bi, cr1);
      ci1 = wmma4(a1r, bi, ci1);  ci1 = wmma4(a1i, br, ci1);
    }
#pragma unroll
    for (int r = 0; r < 8; ++r) {
      const int row0 = wm + r + 8 * half, row1 = row0 + 16;
      WB[0][row0][wn + lr] = cr0[r];  WB[1][row0][wn + lr] = ci0[r];
      WB[0][row1][wn + lr] = cr1[r];  WB[1][row1][wn + lr] = ci1[r];
    }
  }
  __syncthreads();

  // ---- phase 2: WA = WB * GS^T  (low, row-major) ----
  {
    v8f cr0 = {}, cr1 = {}, ci0 = {}, ci1 = {};
#pragma unroll 4
    for (int kk = 0; kk < NPAD; kk += 4) {
      const int kc = kk + 2 * half;
      v2f a0r, a0i, a1r, a1i, br, bi, n0, n1;
      a0r.x = WB[0][wm + lr][kc];       a0r.y = WB[0][wm + lr][kc + 1];
      a0i.x = WB[1][wm + lr][kc];       a0i.y = WB[1][wm + lr][kc + 1];
      a1r.x = WB[0][wm + 16 + lr][kc];  a1r.y = WB[0][wm + 16 + lr][kc + 1];
      a1i.x = WB[1][wm + 16 + lr][kc];  a1i.y = WB[1][wm + 16 + lr][kc + 1];
      br.x = GSr[(wn + lr) * NPAD + kc];  br.y = GSr[(wn + lr) * NPAD + kc + 1];
      bi.x = GSi[(wn + lr) * NPAD + kc];  bi.y = GSi[(wn + lr) * NPAD + kc + 1];
      n0.x = -a0i.x; n0.y = -a0i.y; n1.x = -a1i.x; n1.y = -a1i.y;
      cr0 = wmma4(a0r, br, cr0);  cr0 = wmma4(n0, bi, cr0);
      ci0 = wmma4(a0r, bi, ci0);  ci0 = wmma4(a0i, br, ci0);
      cr1 = wmma4(a1r, br, cr1);  cr1 = wmma4(n1, bi, cr1);
      ci1 = wmma4(a1r, bi, ci1);  ci1 = wmma4(a1i, br, ci1);
    }
    __syncthreads();   // WB reads done before phase-3 overwrites it
#pragma unroll
    for (int r = 0; r < 8; ++r) {
      const int row0 = wm + r + 8 * half, row1 = row0 + 16;
      WA[0][row0][wn + lr] = cr0[r];  WA[1][row0][wn + lr] = ci0[r];
      WA[0][row1][wn + lr] = cr1[r];  WA[1][row1][wn + lr] = ci1[r];
    }
  }
  __syncthreads();

  // ---- phase 3: WB[c][m] = sqrt(Y) * low/|low|  (transposed for next GEMM) ----
  for (int idx = t; idx < PLP; idx += 256) {
    const int m = idx >> 6, c = idx & 63;
    float vr = 0.f, vi = 0.f;
    if (m < NLR && c < NLR) {
      const float re = WA[0][m][c], im = WA[1][m][c];
      const float mag = sqrtf(re * re + im * im);
      float ur = 1.f, ui = 0.f;              // angle(0)=0 -> exp = 1
      if (mag > 0.f) { ur = re / mag; ui = im / mag; }
      const float sy = sqrtf(Y[((size_t)b * LMASK + l) * (NLR * NLR) + m * NLR + c]);
      vr = sy * ur; vi = sy * ui;
    }
    WB[0][c][m] = vr;
    WB[1][c][m] = vi;
  }
  __syncthreads();

  // ---- phase 4: WA = FS * low_new ----
  {
    v8f cr0 = {}, cr1 = {}, ci0 = {}, ci1 = {};
#pragma unroll 4
    for (int kk = 0; kk < NPAD; kk += 4) {
      const int kc = kk + 2 * half;
      v2f a0r, a0i, a1r, a1i, br, bi, n0, n1;
      a0r.x = FSr[(wm + lr) * NPAD + kc];       a0r.y = FSr[(wm + lr) * NPAD + kc + 1];
      a0i.x = FSi[(wm + lr) * NPAD + kc];       a0i.y = FSi[(wm + lr) * NPAD + kc + 1];
      a1r.x = FSr[(wm + 16 + lr) * NPAD + kc];  a1r.y = FSr[(wm + 16 + lr) * NPAD + kc + 1];
      a1i.x = FSi[(wm + 16 + lr) * NPAD + kc];  a1i.y = FSi[(wm + 16 + lr) * NPAD + kc + 1];
      br.x = WB[0][wn + lr][kc];  br.y = WB[0][wn + lr][kc + 1];
      bi.x = WB[1][wn + lr][kc];  bi.y = WB[1][wn + lr][kc + 1];
      n0.x = -a0i.x; n0.y = -a0i.y; n1.x = -a1i.x; n1.y = -a1i.y;
      cr0 = wmma4(a0r, br, cr0);  cr0 = wmma4(n0, bi, cr0);
      ci0 = wmma4(a0r, bi, ci0);  ci0 = wmma4(a0i, br, ci0);
      cr1 = wmma4(a1r, br, cr1);  cr1 = wmma4(n1, bi, cr1);
      ci1 = wmma4(a1r, bi, ci1);  ci1 = wmma4(a1i, br, ci1);
    }
    __syncthreads();   // WA reads of phase-3 consumers done (none) / reuse safety
#pragma unroll
    for (int r = 0; r < 8; ++r) {
      const int row0 = wm + r + 8 * half, row1 = row0 + 16;
      WA[0][row0][wn + lr] = cr0[r];  WA[1][row0][wn + lr] = ci0[r];
      WA[0][row1][wn + lr] = cr1[r];  WA[1][row1][wn + lr] = ci1[r];
    }
  }
  __syncthreads();

  // ---- phase 5: part = WA * FS^T, fused combine + OFT writeback ----
  {
    v8f cr0 = {}, cr1 = {}, ci0 = {}, ci1 = {};
#pragma unroll 4
    for (int kk = 0; kk < NPAD; kk += 4) {
      const int kc = kk + 2 * half;
      v2f a0r, a0i, a1r, a1i, br, bi, n0, n1;
      a0r.x = WA[0][wm + lr][kc];       a0r.y = WA[0][wm + lr][kc + 1];
      a0i.x = WA[1][wm + lr][kc];       a0i.y = WA[1][wm + lr][kc + 1];
      a1r.x = WA[0][wm + 16 + lr][kc];  a1r.y = WA[0][wm + 16 + lr][kc + 1];
      a1i.x = WA[1][wm + 16 + lr][kc];  a1i.y = WA[1][wm + 16 + lr][kc + 1];
      br.x = FSr[(wn + lr) * NPAD + kc];  br.y = FSr[(wn + lr) * NPAD + kc + 1];
      bi.x = FSi[(wn + lr) * NPAD + kc];  bi.y = FSi[(wn + lr) * NPAD + kc + 1];
      n0.x = -a0i.x; n0.y = -a0i.y; n1.x = -a1i.x; n1.y = -a1i.y;
      cr0 = wmma4(a0r, br, cr0);  cr0 = wmma4(n0, bi, cr0);
      ci0 = wmma4(a0r, bi, ci0);  ci0 = wmma4(a0i, br, ci0);
      cr1 = wmma4(a1r, br, cr1);  cr1 = wmma4(n1, bi, cr1);
      ci1 = wmma4(a1r, bi, ci1);  ci1 = wmma4(a1i, br, ci1);
    }
#pragma unroll
    for (int r = 0; r < 8; ++r) {
      const int col = wn + lr;
      const int row0 = wm + r + 8 * half, row1 = row0 + 16;
      if (row0 < NLR && col < NLR) {
        const float ct = CTF[row0 * NLR + col];
        const size_t ad = (size_t)(r0 + row0) * NFULL + c0 + col;
        Or[ad] = (1.f - ct) * Or[ad] + ct * cr0[r];
        Oi[ad] = (1.f - ct) * Oi[ad] + ct * ci0[r];
      }
      if (row1 < NLR && col < NLR) {
        const float ct = CTF[row1 * NLR + col];
        const size_t ad = (size_t)(r0 + row1) * NFULL + c0 + col;
        Or[ad] = (1.f - ct) * Or[ad] + ct * cr1[r];
        Oi[ad] = (1.f - ct) * Oi[ad] + ct * ci1[r];
      }
    }
  }
}

// ---------------------------------------------------------------------------
// Shifted DFT matrix init.
//  forward (fft2s):  F[k][n] = (-1)^n * exp(-2pi i k n / N), scale 1
//  inverse (ifft2s): G[n][k] = (-1)^n * (1/N) * exp(+2pi i n k / N)
// ---------------------------------------------------------------------------
__global__ void k_init_dft(float* __restrict__ M, int Np, int N,
                           float sign, float scale, int shift_row) {
  const int idx = blockIdx.x * blockDim.x + threadIdx.x;
  const int plane = Np * Np;
  if (idx >= plane) return;
  const int k = idx / Np, n = idx - k * Np;
  float re = 0.f, im = 0.f;
  if (k < N && n < N) {
    const int m = (k * n) % N;
    const float ang = sign * 6.28318530717958648f * (float)m / (float)N;
    const float s = (((shift_row ? k : n) & 1) ? -scale : scale);
    re = s * cosf(ang);
    im = s * sinf(ang);
  }
  M[idx] = re;
  M[plane + idx] = im;
}

// ---------------------------------------------------------------------------
// Pointwise / gather / scatter kernels
// ---------------------------------------------------------------------------
__global__ void k_load_real(const float* __restrict__ src, float* __restrict__ dst) {
  const int idx = blockIdx.x * blockDim.x + threadIdx.x;
  if (idx >= BATCH * PLF) return;
  const int b = idx / PLF, p = idx - b * PLF;
  float* d = dst + (size_t)b * 2 * PLF;
  d[p] = src[idx];
  d[PLF + p] = 0.f;
}

__global__ void k_abs(const float* __restrict__ Z, float* __restrict__ A, float clipv) {
  const int idx = blockIdx.x * blockDim.x + threadIdx.x;
  if (idx >= BATCH * PLF) return;
  const int b = idx / PLF, p = idx - b * PLF;
  const float re = Z[(size_t)b * 2 * PLF + p];
  const float im = Z[(size_t)b * 2 * PLF + PLF + p];
  A[idx] = fminf(sqrtf(re * re + im * im), clipv);
}

// a_linear gather: P[b*L+l] = crop(ZF[b], mask_l) * CTF  (64-padded)
__global__ void k_gather_all(const float* __restrict__ ZF, float* __restrict__ P,
                             const float* __restrict__ CTF, const int* __restrict__ Masks) {
  const int idx = blockIdx.x * blockDim.x + threadIdx.x;
  if (idx >= BATCH * LMASK * PLP) return;
  const int f = idx / PLP, p = idx - f * PLP;
  const int b = f / LMASK, l = f - b * LMASK;
  const int i = p >> 6, j = p & 63;
  float vr = 0.f, vi = 0.f;
  if (i < NLR && j < NLR) {
    const int r0 = Masks[2 * l] - 1, c0 = Masks[2 * l + 1] - 1;
    const float* Z = ZF + (size_t)b * 2 * PLF;
    const float c = CTF[i * NLR + j];
    vr = Z[(r0 + i) * NFULL + c0 + j] * c;
    vi = Z[PLF + (r0 + i) * NFULL + c0 + j] * c;
  }
  float* O = P + (size_t)f * 2 * PLP;
  O[p] = vr; O[PLP + p] = vi;
}

// Cz = Bz - sqrt(Y) * exp(i*angle(Bz))
__global__ void k_cz(const float* __restrict__ BZ, float* __restrict__ CZ,
                     const float* __restrict__ Y) {
  const int idx = blockIdx.x * blockDim.x + threadIdx.x;
  if (idx >= BATCH * LMASK * PLP) return;
  const int f = idx / PLP, p = idx - f * PLP;
  const int i = p >> 6, j = p & 63;
  float vr = 0.f, vi = 0.f;
  if (i < NLR && j < NLR) {
    const float* Bw = BZ + (size_t)f * 2 * PLP;
    const float re = Bw[p], im = Bw[PLP + p];
    const float m = sqrtf(re * re + im * im);
    float ur = 1.f, ui = 0.f;
    if (m > 0.f) { ur = re / m; ui = im / m; }
    const float sy = sqrtf(Y[(size_t)f * (NLR * NLR) + i * NLR + j]);
    vr = re - sy * ur; vi = im - sy * ui;
  }
  float* O = CZ + (size_t)f * 2 * PLP;
  O[p] = vr; O[PLP + p] = vi;
}

// a_inverse scatter:  ACC[b, r0+i, c0+j] += XC[b,l,i,j] * CTF   (conj(CTF)=CTF)
__global__ void k_scatter_add(float* __restrict__ ACC, const float* __restrict__ XC,
                              const float* __restrict__ CTF, const int* __restrict__ Masks) {
  const int idx = blockIdx.x * blockDim.x + threadIdx.x;
  if (idx >= BATCH * LMASK * NLR * NLR) return;
  const int f = idx / (NLR * NLR), p = idx - f * (NLR * NLR);
  const int b = f / LMASK, l = f - b * LMASK;
  const int i = p / NLR, j = p - i * NLR;
  const int r0 = Masks[2 * l] - 1, c0 = Masks[2 * l + 1] - 1;
  const float c = CTF[p];
  const float* X = XC + (size_t)f * 2 * PLP;
  const int pp = i * NPAD + j;
  float* A = ACC + (size_t)b * 2 * PLF;
  atomicAdd(&A[(r0 + i) * NFULL + c0 + j],       X[pp] * c);
  atomicAdd(&A[PLF + (r0 + i) * NFULL + c0 + j], X[PLP + pp] * c);
}

// temp_x = img_ra * it / (|it| + eps)
__global__ void k_tempx(const float* __restrict__ RA, const float* __restrict__ IT,
                        float* __restrict__ TX) {
  const int idx = blockIdx.x * blockDim.x + threadIdx.x;
  if (idx >= BATCH * PLF) return;
  const int b = idx / PLF, p = idx - b * PLF;
  const float* I = IT + (size_t)b * 2 * PLF;
  const float re = I[p], im = I[PLF + p];
  const float m = sqrtf(re * re + im * im);
  const float s = RA[idx] / (m + 1e-6f);
  float* T = TX + (size_t)b * 2 * PLF;
  T[p] = s * re; T[PLF + p] = s * im;
}

// z_new = z_f - (10*eta/L)*ACC - (100*eta*lamb)*(z_f - tf)
__global__ void k_znew(const float* __restrict__ ZF, const float* __restrict__ ZUPD,
                       const float* __restrict__ TF, float* __restrict__ ZN,
                       const float* __restrict__ lamb, const float* __restrict__ eta1) {
  const int idx = blockIdx.x * blockDim.x + threadIdx.x;
  if (idx >= BATCH * PLF) return;
  const int b = idx / PLF, p = idx - b * PLF;
  const float eta = eta1[0], lam = lamb[0];
  const float c1 = 10.f * eta / (float)LMASK;
  const float c2 = 100.f * eta * lam;
  const size_t b2 = (size_t)b * 2 * PLF;
  {
    const float z = ZF[b2 + p], u = ZUPD[b2 + p], tt = TF[b2 + p];
    ZN[b2 + p] = z - c1 * u - c2 * (z - tt);
  }
  {
    const float z = ZF[b2 + PLF + p], u = ZUPD[b2 + PLF + p], tt = TF[b2 + PLF + p];
    ZN[b2 + PLF + p] = z - c1 * u - c2 * (z - tt);
  }
}

// outputs: img_ra (B*65536 f32) then img_rc as interleaved complex64
__global__ void k_output(const float* __restrict__ RA, const float* __restrict__ RC,
                         float* __restrict__ out) {
  const int idx = blockIdx.x * blockDim.x + threadIdx.x;
  if (idx >= BATCH * PLF) return;
  const int b = idx / PLF, p = idx - b * PLF;
  out[idx] = RA[idx];
  const float* R = RC + (size_t)b * 2 * PLF;
  const size_t ofs = (size_t)BATCH * PLF;
  out[ofs + 2 * (size_t)idx]     = R[p];
  out[ofs + 2 * (size_t)idx + 1] = R[PLF + p];
}

// ---------------------------------------------------------------------------
// Host orchestration
// ---------------------------------------------------------------------------
static inline void cgemm_MX(const float* M, const float* X, float* C, int Np, int batch,
                            hipStream_t s) {
  dim3 g(Np / 64, Np / 64, batch), blk(128);
  k_cgemm<0><<<g, blk, 0, s>>>(M, X, C, Np, 0ull,
                               (unsigned long long)(2 * Np * Np));
}
static inline void cgemm_XMT(const float* X, const float* M, float* C, int Np, int batch,
                             hipStream_t s) {
  dim3 g(Np / 64, Np / 64, batch), blk(128);
  k_cgemm<1><<<g, blk, 0, s>>>(X, M, C, Np,
                               (unsigned long long)(2 * Np * Np), 0ull);
}

extern "C" void kernel_launch(void* const* d_in, const int* in_sizes, int n_in,
                              void* d_out, int out_size, void* d_ws, size_t ws_size,
                              hipStream_t stream) {
  (void)in_sizes; (void)n_in; (void)out_size; (void)ws_size;
  const float* Img   = (const float*)d_in[0];
  const float* Y     = (const float*)d_in[1];
  const float* CTF   = (const float*)d_in[2];
  const float* lamb  = (const float*)d_in[3];
  const float* eta1  = (const float*)d_in[4];
  const int*   Masks = (const int*)d_in[5];
  float* out = (float*)d_out;

  float* ws = (float*)d_ws;
  size_t off = 0;
  auto alloc = [&](size_t n) { float* p = ws + off; off += n; return p; };

  float* FS256 = alloc(2 * PLF);                  // shifted fwd DFT, 256
  float* GS256 = alloc(2 * PLF);                  // shifted inv DFT, 256
  float* FS64  = alloc(2 * PLP);                  // shifted fwd DFT, 50 in 64-pad
  float* GS64  = alloc(2 * PLP);                  // shifted inv DFT, 50 in 64-pad
  float* OFT   = alloc((size_t)BATCH * 2 * PLF);
  float* TA    = alloc((size_t)BATCH * 2 * PLF);
  float* TB    = alloc((size_t)BATCH * 2 * PLF);
  float* ZF    = alloc((size_t)BATCH * 2 * PLF);
  float* ZUPD  = alloc((size_t)BATCH * 2 * PLF);
  float* IT    = alloc((size_t)BATCH * 2 * PLF);
  float* TX    = alloc((size_t)BATCH * 2 * PLF);
  float* IMRC  = alloc((size_t)BATCH * 2 * PLF);
  float* IMRA  = alloc((size_t)BATCH * PLF);
  float* P50A  = alloc((size_t)BATCH * LMASK * 2 * PLP);
  float* P50B  = alloc((size_t)BATCH * LMASK * 2 * PLP);
  float* P50C  = alloc((size_t)BATCH * LMASK * 2 * PLP);

  const int grF  = (BATCH * PLF + 255) / 256;
  const int grG  = (BATCH * LMASK * PLP + 255) / 256;
  const int grSA = (BATCH * LMASK * NLR * NLR + 255) / 256;

  // --- DFT matrices (fftshift/ifftshift folded in as (-1)^index prefactors) ---
  k_init_dft<<<(PLF + 255) / 256, 256, 0, stream>>>(FS256, NFULL, NFULL, -1.f, 1.f, 0);
  k_init_dft<<<(PLF + 255) / 256, 256, 0, stream>>>(GS256, NFULL, NFULL, +1.f, 1.f / NFULL, 1);
  k_init_dft<<<(PLP + 255) / 256, 256, 0, stream>>>(FS64, NPAD, NLR, -1.f, 1.f, 0);
  k_init_dft<<<(PLP + 255) / 256, 256, 0, stream>>>(GS64, NPAD, NLR, +1.f, 1.f / NLR, 1);

  // ---------------- fp_basic ----------------
  k_load_real<<<grF, 256, 0, stream>>>(Img, TA);
  cgemm_MX(FS256, TA, TB, NFULL, BATCH, stream);       // OFT = fft2s(img)
  cgemm_XMT(TB, FS256, OFT, NFULL, BATCH, stream);

  for (int t = 0; t < TSWEEP; ++t) {
    for (int l = 0; l < LMASK; ++l) {
      k_fp_step<<<dim3(BATCH), dim3(256), 0, stream>>>(OFT, GS64, FS64, CTF, Y, Masks, l);
    }
  }
  cgemm_MX(GS256, OFT, TA, NFULL, BATCH, stream);      // im_rc = ifft2s(OFT)
  cgemm_XMT(TA, GS256, IMRC, NFULL, BATCH, stream);
  k_abs<<<grF, 256, 0, stream>>>(IMRC, IMRA, 100.f);   // img_ra = clip(|im_rc|,0,100)

  // ---------------- AFFP x 8 ----------------
  for (int s = 0; s < SITER; ++s) {
    cgemm_MX(FS256, IMRC, TA, NFULL, BATCH, stream);   // z_f = fft2s(img_rc)
    cgemm_XMT(TA, FS256, ZF, NFULL, BATCH, stream);

    k_gather_all<<<grG, 256, 0, stream>>>(ZF, P50A, CTF, Masks);   // a_linear
    cgemm_MX(GS64, P50A, P50B, NPAD, BATCH * LMASK, stream);
    cgemm_XMT(P50B, GS64, P50C, NPAD, BATCH * LMASK, stream);      // Bz

    k_cz<<<grG, 256, 0, stream>>>(P50C, P50A, Y);                  // Cz

    cgemm_MX(FS64, P50A, P50B, NPAD, BATCH * LMASK, stream);       // a_inverse fft
    cgemm_XMT(P50B, FS64, P50C, NPAD, BATCH * LMASK, stream);
    hipMemsetAsync(ZUPD, 0, (size_t)BATCH * 2 * PLF * sizeof(float), stream);
    k_scatter_add<<<grSA, 256, 0, stream>>>(ZUPD, P50C, CTF, Masks);

    cgemm_MX(GS256, ZF, TA, NFULL, BATCH, stream);     // it = ifft2s(z_f)
    cgemm_XMT(TA, GS256, IT, NFULL, BATCH, stream);
    k_tempx<<<grF, 256, 0, stream>>>(IMRA, IT, TX);    // temp_x

    cgemm_MX(FS256, TX, TA, NFULL, BATCH, stream);     // tf = fft2s(temp_x)
    cgemm_XMT(TA, FS256, TB, NFULL, BATCH, stream);

    k_znew<<<grF, 256, 0, stream>>>(ZF, ZUPD, TB, TX, lamb, eta1);  // z_new -> TX

    cgemm_MX(GS256, TX, TA, NFULL, BATCH, stream);     // im_rc = ifft2s(z_new)
    cgemm_XMT(TA, GS256, IMRC, NFULL, BATCH, stream);
    k_abs<<<grF, 256, 0, stream>>>(IMRC, IMRA, 1e30f); // img_ra = |im_rc|
  }

  k_output<<<grF, 256, 0, stream>>>(IMRA, IMRC, out);
}